// BandletTransform3D_16844861735717
// MI455X (gfx1250) — compile-verified
//
#include <hip/hip_runtime.h>

// ---------------------------------------------------------------------------
// BandletTransform3D for MI455X (gfx1250, wave32).
// f32 end-to-end; 2D Haar on 8x8 slices done via V_WMMA_F32_16X16X4_F32 on
// 16x16 tiles holding four 8x8 slices with block-diagonal Hd = diag(H8,H8).
// ---------------------------------------------------------------------------

typedef float v2f __attribute__((ext_vector_type(2)));
typedef float v8f __attribute__((ext_vector_type(8)));

#define INV_SQRT8 0.3535533905932738f
#define SQH 0.7071067811865476f
#define TAU_ 0.05f

// 8x8 orthonormal Haar matrix H (Mallat ordering), row-major.
__constant__ float H8c[64] = {
  INV_SQRT8, INV_SQRT8, INV_SQRT8, INV_SQRT8, INV_SQRT8, INV_SQRT8, INV_SQRT8, INV_SQRT8,
  INV_SQRT8, INV_SQRT8, INV_SQRT8, INV_SQRT8,-INV_SQRT8,-INV_SQRT8,-INV_SQRT8,-INV_SQRT8,
  0.5f, 0.5f,-0.5f,-0.5f, 0.0f, 0.0f, 0.0f, 0.0f,
  0.0f, 0.0f, 0.0f, 0.0f, 0.5f, 0.5f,-0.5f,-0.5f,
  SQH ,-SQH , 0.0f, 0.0f, 0.0f, 0.0f, 0.0f, 0.0f,
  0.0f, 0.0f, SQH ,-SQH , 0.0f, 0.0f, 0.0f, 0.0f,
  0.0f, 0.0f, 0.0f, 0.0f, SQH ,-SQH , 0.0f, 0.0f,
  0.0f, 0.0f, 0.0f, 0.0f, 0.0f, 0.0f, SQH ,-SQH
};

// Hd = diag(H8, H8) as a 16x16 matrix. Branchless select.
__device__ __forceinline__ float hd16(int r, int c) {
  float v = H8c[((r & 7) << 3) | (c & 7)];
  return (((r >> 3) ^ (c >> 3)) & 1) ? 0.0f : v;
}

// (slice, row-in-slice, col-in-slice) -> d*64+h*8+w in an 8^3 block,
// per analysis normal n. n is a compile-time constant after unrolling.
__device__ __forceinline__ int blkIndex(int n, int sl, int rr, int cc) {
  if (n == 0)      return (sl << 6) + (rr << 3) + cc;   // d=sl h=rr w=cc
  else if (n == 1) return (rr << 6) + (sl << 3) + cc;   // d=rr h=sl w=cc
  else             return (rr << 6) + (cc << 3) + sl;   // d=rr h=cc w=sl
}

// Index of element (rowX, colX) of the 16x16 tile t (quadrants = 4 slices).
__device__ __forceinline__ int xidx(int n, int t, int rowX, int colX) {
  int sl = t * 4 + ((rowX >> 3) << 1) + (colX >> 3);
  return blkIndex(n, sl, rowX & 7, colX & 7);
}

// Branchless soft threshold: copysign(max(|v|-tau, 0), v).
__device__ __forceinline__ float softThresh(float v) {
  float av = fmaxf(fabsf(v) - TAU_, 0.0f);
  return copysignf(av, v);
}

// ---------------------------------------------------------------------------
// Kernel A: one forward 3D Haar level. in: [nB][D][H][W]  out: 8 bands
// [band][nB][D/2][H/2][W/2], band = sd*4+sh*2+sw.
// ---------------------------------------------------------------------------
__global__ __launch_bounds__(256)
void dwt_fwd(const float* __restrict__ in, float* __restrict__ out,
             int D, int H, int W, int nB) {
  int Dc = D >> 1, Hc = H >> 1, Wc = W >> 1;
  size_t total = (size_t)nB * Dc * Hc * Wc;
  size_t tid = (size_t)blockIdx.x * blockDim.x + threadIdx.x;
  if (tid >= total) return;
  int x = (int)(tid % Wc); size_t r = tid / Wc;
  int y = (int)(r % Hc);   r /= Hc;
  int z = (int)(r % Dc);   int b = (int)(r / Dc);

  const float* p0 = in + ((((size_t)b * D + 2 * z) * H + 2 * y) * W + 2 * x);
  const float* p1 = p0 + (size_t)H * W;
  float a000 = p0[0], a001 = p0[1], a010 = p0[W], a011 = p0[W + 1];
  float a100 = p1[0], a101 = p1[1], a110 = p1[W], a111 = p1[W + 1];

  float w0_00 = a000 + a001, w1_00 = a000 - a001;
  float w0_01 = a010 + a011, w1_01 = a010 - a011;
  float w0_10 = a100 + a101, w1_10 = a100 - a101;
  float w0_11 = a110 + a111, w1_11 = a110 - a111;
  float h00_0 = w0_00 + w0_01, h00_1 = w0_10 + w0_11;
  float h01_0 = w1_00 + w1_01, h01_1 = w1_10 + w1_11;
  float h10_0 = w0_00 - w0_01, h10_1 = w0_10 - w0_11;
  float h11_0 = w1_00 - w1_01, h11_1 = w1_10 - w1_11;

  const float s = INV_SQRT8;
  float v[8];
  v[0] = (h00_0 + h00_1) * s;  v[1] = (h01_0 + h01_1) * s;
  v[2] = (h10_0 + h10_1) * s;  v[3] = (h11_0 + h11_1) * s;
  v[4] = (h00_0 - h00_1) * s;  v[5] = (h01_0 - h01_1) * s;
  v[6] = (h10_0 - h10_1) * s;  v[7] = (h11_0 - h11_1) * s;

  size_t cIdx = (((size_t)b * Dc + z) * Hc + y) * Wc + x;
  size_t bs = (size_t)nB * Dc * Hc * Wc;
#pragma unroll
  for (int bd = 0; bd < 8; ++bd) out[bd * bs + cIdx] = v[bd];
}

// ---------------------------------------------------------------------------
// Kernel C: one inverse 3D Haar level. bands: [8][nB][Dc][Hc][Wc] ->
// out volume [nB][2Dc][2Hc][2Wc].
// ---------------------------------------------------------------------------
__global__ __launch_bounds__(256)
void dwt_inv(const float* __restrict__ bands, float* __restrict__ out,
             int Dc, int Hc, int Wc, int nB) {
  size_t total = (size_t)nB * Dc * Hc * Wc;
  size_t tid = (size_t)blockIdx.x * blockDim.x + threadIdx.x;
  if (tid >= total) return;
  int x = (int)(tid % Wc); size_t r = tid / Wc;
  int y = (int)(r % Hc);   r /= Hc;
  int z = (int)(r % Dc);   int b = (int)(r / Dc);

  size_t cIdx = (((size_t)b * Dc + z) * Hc + y) * Wc + x;
  size_t bs = (size_t)nB * Dc * Hc * Wc;
  float c[8];
#pragma unroll
  for (int bd = 0; bd < 8; ++bd) c[bd] = bands[bd * bs + cIdx];

  float g00_0 = c[0] + c[4], g00_1 = c[0] - c[4];
  float g01_0 = c[1] + c[5], g01_1 = c[1] - c[5];
  float g10_0 = c[2] + c[6], g10_1 = c[2] - c[6];
  float g11_0 = c[3] + c[7], g11_1 = c[3] - c[7];
  float f0_00 = g00_0 + g10_0, f0_01 = g00_0 - g10_0;
  float f0_10 = g00_1 + g10_1, f0_11 = g00_1 - g10_1;
  float f1_00 = g01_0 + g11_0, f1_01 = g01_0 - g11_0;
  float f1_10 = g01_1 + g11_1, f1_11 = g01_1 - g11_1;

  const float s = INV_SQRT8;
  int D = 2 * Dc, H = 2 * Hc, W = 2 * Wc;
  float* p0 = out + ((((size_t)b * D + 2 * z) * H + 2 * y) * W + 2 * x);
  float* p1 = p0 + (size_t)H * W;
  p0[0]     = (f0_00 + f1_00) * s;  p0[1]     = (f0_00 - f1_00) * s;
  p0[W]     = (f0_01 + f1_01) * s;  p0[W + 1] = (f0_01 - f1_01) * s;
  p1[0]     = (f0_10 + f1_10) * s;  p1[1]     = (f0_10 - f1_10) * s;
  p1[W]     = (f0_11 + f1_11) * s;  p1[W + 1] = (f0_11 - f1_11) * s;
}

// ---------------------------------------------------------------------------
// Kernel B: bandlet processing of one 8^3 block per wave32 workgroup,
// in place on band buffers bands[8][nB][S][S][S] (bands 1..7).
// Per normal n and 16x16 tile t: Cc = Hd X Hd^T, soft-threshold (preserve
// per-slice DC), R = Hd^T Ct Hd, accumulate; finally write acc/3.
// Fully unrolled: 3 normals x 2 tiles x 4 matmuls x 4 K-chunks = 96 WMMA.
// ---------------------------------------------------------------------------
__global__ __launch_bounds__(32)
void bandlet_k(float* __restrict__ bands, int S, int nb, int nB) {
  __shared__ __align__(16) float blk[512];
  __shared__ __align__(16) float acc[512];
  __shared__ __align__(16) float tmp[256];

  const int lane = threadIdx.x;     // 0..31, wave32
  const int m16  = lane & 15;
  const int hi   = lane >> 4;

  int wg = blockIdx.x;
  int bpb = nB * nb * nb * nb;      // blocks per band
  int bd = 1 + wg / bpb;
  int r  = wg % bpb;
  int b  = r / (nb * nb * nb); r %= nb * nb * nb;
  int bz = r / (nb * nb);      r %= nb * nb;
  int by = r / nb;
  int bx = r - by * nb;

  size_t bandStride = (size_t)nB * S * S * S;
  float* g = bands + (size_t)bd * bandStride
           + ((((size_t)b * S + bz * 8) * S + by * 8) * S + bx * 8);

  // Vector load of the 8^3 block: 4 x float4 per lane (16B aligned: all
  // offsets are multiples of 8 floats since S is a multiple of 8).
#pragma unroll
  for (int i = 0; i < 4; ++i) {
    int idx = (i * 32 + lane) * 4;              // multiple of 4
    int d = idx >> 6, h = (idx >> 3) & 7, w = idx & 7;  // w in {0,4}
    float4 val = *(const float4*)(g + ((size_t)d * S + h) * S + w);
    *(float4*)&blk[idx] = val;
    *(float4*)&acc[idx] = make_float4(0.f, 0.f, 0.f, 0.f);
  }
  __syncthreads();

  // Per-lane constant WMMA fragments of Hd, reused by all matmuls:
  //  hA[k] = {Hd[m16][c0], Hd[m16][c0+1]}  : A-frag of Hd,  B-frag of HdT
  //  hB[k] = {Hd[c0][m16], Hd[c0+1][m16]}  : A-frag of HdT, B-frag of Hd
  v2f hA[4], hB[4];
#pragma unroll
  for (int k = 0; k < 4; ++k) {
    int c0 = 4 * k + 2 * hi;
    hA[k].x = hd16(m16, c0);     hA[k].y = hd16(m16, c0 + 1);
    hB[k].x = hd16(c0, m16);     hB[k].y = hd16(c0 + 1, m16);
  }

#pragma unroll
  for (int n = 0; n < 3; ++n) {
#pragma unroll
    for (int t = 0; t < 2; ++t) {
      // ---- T = Hd * X ----  (B-frags gathered from blk)
      v8f cT = {0,0,0,0,0,0,0,0};
#pragma unroll
      for (int k = 0; k < 4; ++k) {
        int c0 = 4 * k + 2 * hi;
        v2f bf;
        bf.x = blk[xidx(n, t, c0,     m16)];
        bf.y = blk[xidx(n, t, c0 + 1, m16)];
        cT = __builtin_amdgcn_wmma_f32_16x16x4_f32(false, hA[k], false, bf,
                                                   (short)0, cT, false, false);
      }
#pragma unroll
      for (int i = 0; i < 8; ++i) tmp[(i + 8 * hi) * 16 + m16] = cT[i];
      __syncthreads();

      // ---- Cc = T * Hd^T ----  (A from tmp, B = hA)
      v8f cC = {0,0,0,0,0,0,0,0};
#pragma unroll
      for (int k = 0; k < 4; ++k) {
        int c0 = 4 * k + 2 * hi;
        v2f a;
        a.x = tmp[m16 * 16 + c0];
        a.y = tmp[m16 * 16 + c0 + 1];
        cC = __builtin_amdgcn_wmma_f32_16x16x4_f32(false, a, false, hA[k],
                                                   (short)0, cC, false, false);
      }
      __syncthreads();               // everyone done reading T from tmp

      // ---- soft threshold (preserve per-slice DC at (0,0) of each 8x8) ----
      // mm & 7 == i (mm = i + 8*hi, i < 8), so only i == 0 can be a DC
      // element; that case reduces to a single cndmask on (nn & 7) == 0.
#pragma unroll
      for (int i = 0; i < 8; ++i) {
        int mm = i + 8 * hi, nn = m16;
        float v  = cC[i];
        float sv = softThresh(v);
        if (i == 0) v = ((nn & 7) == 0) ? v : sv;   // i literal: folds
        else        v = sv;
        tmp[mm * 16 + nn] = v;
      }
      __syncthreads();

      // ---- U = Hd^T * Ct ----  (A = hB, B from tmp)
      v8f cU = {0,0,0,0,0,0,0,0};
#pragma unroll
      for (int k = 0; k < 4; ++k) {
        int c0 = 4 * k + 2 * hi;
        v2f bf;
        bf.x = tmp[c0 * 16 + m16];
        bf.y = tmp[(c0 + 1) * 16 + m16];
        cU = __builtin_amdgcn_wmma_f32_16x16x4_f32(false, hB[k], false, bf,
                                                   (short)0, cU, false, false);
      }
      __syncthreads();
#pragma unroll
      for (int i = 0; i < 8; ++i) tmp[(i + 8 * hi) * 16 + m16] = cU[i];
      __syncthreads();

      // ---- R = U * Hd ----  (A from tmp, B = hB)
      v8f cR = {0,0,0,0,0,0,0,0};
#pragma unroll
      for (int k = 0; k < 4; ++k) {
        int c0 = 4 * k + 2 * hi;
        v2f a;
        a.x = tmp[m16 * 16 + c0];
        a.y = tmp[m16 * 16 + c0 + 1];
        cR = __builtin_amdgcn_wmma_f32_16x16x4_f32(false, a, false, hB[k],
                                                   (short)0, cR, false, false);
      }
      // accumulate into block accumulator (distinct LDS addrs per lane)
#pragma unroll
      for (int i = 0; i < 8; ++i) {
        int mm = i + 8 * hi, nn = m16;
        int sl = t * 4 + ((mm >> 3) << 1) + (nn >> 3);
        acc[blkIndex(n, sl, mm & 7, nn & 7)] += cR[i];
      }
      __syncthreads();
    }
  }

  // Vector store of acc/3.
#pragma unroll
  for (int i = 0; i < 4; ++i) {
    int idx = (i * 32 + lane) * 4;
    int d = idx >> 6, h = (idx >> 3) & 7, w = idx & 7;
    float4 val = *(float4*)&acc[idx];
    val.x *= (1.0f / 3.0f); val.y *= (1.0f / 3.0f);
    val.z *= (1.0f / 3.0f); val.w *= (1.0f / 3.0f);
    *(float4*)(g + ((size_t)d * S + h) * S + w) = val;
  }
}

// ---------------------------------------------------------------------------
// Host-side orchestration (graph-capture safe: launches only).
// ws layout: bands1[8][2][80^3] then bands2[8][2][40^3]  (~37 MB of f32).
// ---------------------------------------------------------------------------
extern "C" void kernel_launch(void* const* d_in, const int* in_sizes, int n_in,
                              void* d_out, int out_size, void* d_ws, size_t ws_size,
                              hipStream_t stream) {
  (void)in_sizes; (void)n_in; (void)out_size; (void)ws_size;
  const float* x = (const float*)d_in[0];
  float* out = (float*)d_out;
  float* ws = (float*)d_ws;

  const int nB = 2;
  float* bands1 = ws;                                      // 8*2*80^3 floats
  float* bands2 = ws + (size_t)8 * nB * 80 * 80 * 80;      // 8*2*40^3 floats

  // Forward level 1: x (160^3) -> bands1 (80^3 x 8)
  {
    size_t total = (size_t)nB * 80 * 80 * 80;
    dwt_fwd<<<(unsigned)((total + 255) / 256), 256, 0, stream>>>(
        x, bands1, 160, 160, 160, nB);
  }
  // Forward level 2: bands1 band0 (lll, 80^3) -> bands2 (40^3 x 8)
  {
    size_t total = (size_t)nB * 40 * 40 * 40;
    dwt_fwd<<<(unsigned)((total + 255) / 256), 256, 0, stream>>>(
        bands1, bands2, 80, 80, 80, nB);
  }
  // Bandlet processing (in place) on detail bands, both levels.
  bandlet_k<<<7 * nB * 5 * 5 * 5, 32, 0, stream>>>(bands2, 40, 5, nB);
  bandlet_k<<<7 * nB * 10 * 10 * 10, 32, 0, stream>>>(bands1, 80, 10, nB);
  // Inverse level 2: bands2 (band0 = approx) -> overwrite bands1 band0 slot.
  {
    size_t total = (size_t)nB * 40 * 40 * 40;
    dwt_inv<<<(unsigned)((total + 255) / 256), 256, 0, stream>>>(
        bands2, bands1, 40, 40, 40, nB);
  }
  // Inverse level 1: bands1 -> d_out (160^3).
  {
    size_t total = (size_t)nB * 80 * 80 * 80;
    dwt_inv<<<(unsigned)((total + 255) / 256), 256, 0, stream>>>(
        bands1, out, 80, 80, 80, nB);
  }
}